// HGNNLayer_87256555585991
// MI455X (gfx1250) — compile-verified
//
#include <hip/hip_runtime.h>
#include <math.h>

#define B_   32
#define NU_  256
#define NT_  512
#define H_   8
#define NEGV (-1e9f)
#define EPS_ (1e-5f)

typedef __attribute__((ext_vector_type(2))) float v2f;
typedef __attribute__((ext_vector_type(8))) float v8f;

// ---------------------------------------------------------------------------
// Kernel 1: projections  usv_q = usv@Wq+bq ; task_k = task@Wk+bk ; t_q = task@Wq+bq
// ---------------------------------------------------------------------------
__global__ __launch_bounds__(256) void hgnn_proj_kernel(
    const float* __restrict__ usv, const float* __restrict__ task,
    const float* __restrict__ Wq, const float* __restrict__ bq,
    const float* __restrict__ Wk, const float* __restrict__ bk,
    float* __restrict__ usv_q, float* __restrict__ task_k, float* __restrict__ t_q)
{
    int i = blockIdx.x * blockDim.x + threadIdx.x;
    const int nU = B_ * NU_, nT = B_ * NT_;
    if (i < nU) {
        float xv[H_];
        #pragma unroll
        for (int h = 0; h < H_; h++) xv[h] = usv[(size_t)i * H_ + h];
        #pragma unroll
        for (int o = 0; o < H_; o++) {
            float s = bq[o];
            #pragma unroll
            for (int h = 0; h < H_; h++) s += xv[h] * Wq[h * H_ + o];
            usv_q[(size_t)i * H_ + o] = s;
        }
    } else if (i < nU + nT) {
        int j = i - nU;
        float xv[H_];
        #pragma unroll
        for (int h = 0; h < H_; h++) xv[h] = task[(size_t)j * H_ + h];
        #pragma unroll
        for (int o = 0; o < H_; o++) {
            float sk = bk[o], sq = bq[o];
            #pragma unroll
            for (int h = 0; h < H_; h++) { sk += xv[h] * Wk[h * H_ + o]; sq += xv[h] * Wq[h * H_ + o]; }
            task_k[(size_t)j * H_ + o] = sk;
            t_q  [(size_t)j * H_ + o] = sq;
        }
    }
}

// ---------------------------------------------------------------------------
// Kernel 2: block 1 fused — usv attends to tasks with rank-1 edge bias.
// One block = 16 usv rows x all 512 tasks. Scores->softmax in LDS, then
// aggregation weights[16x512] @ task_embed[512x8(pad16)] via WMMA f32 16x16x4
// (transposed LDS embedding: contiguous b64 fragment loads, cndmask padding),
// then ELU + LayerNorm -> usv_new.
// ---------------------------------------------------------------------------
__global__ __launch_bounds__(64) void hgnn_block1_kernel(
    const float* __restrict__ usv, const float* __restrict__ task,
    const int* __restrict__ adj, const float* __restrict__ ef,
    const float* __restrict__ We, const float* __restrict__ be,
    const float* __restrict__ Wau, const float* __restrict__ bau,
    const float* __restrict__ gnu_, const float* __restrict__ bnu_,
    const float* __restrict__ usv_q, const float* __restrict__ task_k,
    float* __restrict__ usv_new)
{
    __shared__ float w_s[16][NT_];     // 32 KB: scores -> softmax weights (A matrix)
    __shared__ float te_t[H_][NT_];    // 16 KB: task_embed transposed (B matrix source)
    __shared__ float q_s[16][H_];
    __shared__ float agg_s[16][16];

    const int tid  = threadIdx.x;
    const int lane = tid & 31;
    const int wv   = tid >> 5;
    const int b    = blockIdx.x / (NU_ / 16);
    const int u0   = (blockIdx.x % (NU_ / 16)) * 16;

    float WeR[H_], beR[H_], WauR[H_];
    #pragma unroll
    for (int h = 0; h < H_; h++) { WeR[h] = We[h]; beR[h] = be[h]; WauR[h] = Wau[h]; }
    const float bau0 = bau[0];

    // transpose task_embed into LDS: te_t[h][t]
    for (int i = tid; i < NT_ * H_; i += 64) {
        float v = task[(size_t)b * NT_ * H_ + i];
        int t = i >> 3, h = i & (H_ - 1);
        te_t[h][t] = v;
    }
    for (int i = tid; i < 16 * H_; i += 64)
        ((float*)q_s)[i] = usv_q[((size_t)b * NU_ + u0) * H_ + i];
    __syncthreads();

    // ---- scores (with mask) ----
    for (int idx = tid; idx < 16 * NT_; idx += 64) {
        int r = idx >> 9, t = idx & (NT_ - 1);
        size_t g = ((size_t)(b * NU_ + u0 + r)) * NT_ + t;
        float  e = ef[g];
        int    a = adj[g];
        const float* kk = task_k + ((size_t)b * NT_ + t) * H_;
        float s = bau0;
        #pragma unroll
        for (int h = 0; h < H_; h++)
            s += tanhf(q_s[r][h] + kk[h] + e * WeR[h] + beR[h]) * WauR[h];
        w_s[r][t] = (a == 0) ? NEGV : s;
    }
    __syncthreads();

    // ---- softmax per row (wave wv handles rows wv, wv+2, ...) ----
    for (int r = wv; r < 16; r += 2) {
        float m = -INFINITY;
        for (int t = lane; t < NT_; t += 32) m = fmaxf(m, w_s[r][t]);
        #pragma unroll
        for (int o = 16; o; o >>= 1) m = fmaxf(m, __shfl_xor(m, o, 32));
        float l = 0.f;
        for (int t = lane; t < NT_; t += 32) { float v = expf(w_s[r][t] - m); w_s[r][t] = v; l += v; }
        #pragma unroll
        for (int o = 16; o; o >>= 1) l += __shfl_xor(l, o, 32);
        float inv = 1.0f / l;
        for (int t = lane; t < NT_; t += 32) w_s[r][t] *= inv;
    }
    __syncthreads();

    // ---- aggregation via V_WMMA_F32_16X16X4_F32, K-loop over 512 ----
    if (wv == 0) {
        const int  half   = lane >> 4;     // K half: lanes 0-15 -> K0,K1 ; 16-31 -> K2,K3
        const int  l16    = lane & 15;
        const bool nvalid = (l16 < H_);    // N-column validity (8..15 are zero padding)
        const int  bn     = l16 & (H_ - 1);
        v8f c = {0.f, 0.f, 0.f, 0.f, 0.f, 0.f, 0.f, 0.f};
        for (int k0 = 0; k0 < NT_; k0 += 4) {
            int ka = k0 + half * 2;
            float ax = w_s[l16][ka];           // A[M=l16][K=ka]
            float ay = w_s[l16][ka + 1];       // A[M=l16][K=ka+1]  (contiguous b64)
            float tx = te_t[bn][ka];           // B[K=ka][N=l16]    (contiguous b64)
            float ty = te_t[bn][ka + 1];
            v2f a, bb;
            a.x  = ax;
            a.y  = ay;
            bb.x = nvalid ? tx : 0.f;          // pure v_cndmask, no exec masking
            bb.y = nvalid ? ty : 0.f;
            c = __builtin_amdgcn_wmma_f32_16x16x4_f32(false, a, false, bb, (short)0, c, false, false);
        }
        #pragma unroll
        for (int j = 0; j < 8; j++) agg_s[half * 8 + j][l16] = c[j];  // D[M=half*8+j][N=l16]
    }
    __syncthreads();

    // ---- ELU + LayerNorm -> usv_new ----
    if (tid < 16) {
        int r = tid;
        size_t base = ((size_t)b * NU_ + u0 + r) * H_;
        float x[H_]; float mu = 0.f;
        #pragma unroll
        for (int h = 0; h < H_; h++) {
            float v = usv[base + h] + agg_s[r][h];
            v = (v > 0.f) ? v : (expf(v) - 1.f);
            x[h] = v; mu += v;
        }
        mu *= (1.0f / H_);
        float var = 0.f;
        #pragma unroll
        for (int h = 0; h < H_; h++) { float d = x[h] - mu; var += d * d; }
        var *= (1.0f / H_);
        float rs = rsqrtf(var + EPS_);
        #pragma unroll
        for (int h = 0; h < H_; h++)
            usv_new[base + h] = (x[h] - mu) * rs * gnu_[h] + bnu_[h];
    }
}

// ---------------------------------------------------------------------------
// Kernel 3: KNN top-3 (on squared distance; sqrt is monotone so ordering and
// ties are identical) + 3-way masked softmax + agg_nbr.  NEG masking makes
// the dense NTxNT softmax exactly equal to softmax over the 3 neighbors.
// ---------------------------------------------------------------------------
__global__ __launch_bounds__(256) void hgnn_knn_kernel(
    const float* __restrict__ task, const float* __restrict__ coords,
    const float* __restrict__ t_q, const float* __restrict__ task_k,
    const float* __restrict__ Wat, const float* __restrict__ bat,
    float* __restrict__ agg_nbr)
{
    int i = blockIdx.x * blockDim.x + threadIdx.x;
    if (i >= B_ * NT_) return;
    int b = i / NT_;
    int t = i - b * NT_;
    const float* cb = coords + (size_t)b * NT_ * 2;
    float cx = cb[t * 2 + 0], cy = cb[t * 2 + 1];

    float d0 = 3.4e38f, d1 = 3.4e38f, d2 = 3.4e38f;
    int   i0 = 0, i1 = 0, i2 = 0;
    for (int s = 0; s < NT_; s++) {
        float dx = cx - cb[s * 2 + 0];
        float dy = cy - cb[s * 2 + 1];
        float d  = dx * dx + dy * dy;
        if (d < d0)      { d2 = d1; i2 = i1; d1 = d0; i1 = i0; d0 = d; i0 = s; }
        else if (d < d1) { d2 = d1; i2 = i1; d1 = d;  i1 = s; }
        else if (d < d2) { d2 = d;  i2 = s; }
    }

    const float* tq = t_q + (size_t)i * H_;
    float Wt[H_];
    #pragma unroll
    for (int h = 0; h < H_; h++) Wt[h] = Wat[h];
    const float bt = bat[0];

    int nb[3] = { i0, i1, i2 };
    float sc[3];
    #pragma unroll
    for (int j = 0; j < 3; j++) {
        const float* kk = task_k + ((size_t)b * NT_ + nb[j]) * H_;
        float s = bt;
        #pragma unroll
        for (int h = 0; h < H_; h++) s += tanhf(tq[h] + kk[h]) * Wt[h];
        sc[j] = s;
    }
    float m  = fmaxf(sc[0], fmaxf(sc[1], sc[2]));
    float e0 = expf(sc[0] - m), e1 = expf(sc[1] - m), e2 = expf(sc[2] - m);
    float inv = 1.0f / (e0 + e1 + e2);
    float w[3] = { e0 * inv, e1 * inv, e2 * inv };

    float out[H_] = {0.f, 0.f, 0.f, 0.f, 0.f, 0.f, 0.f, 0.f};
    #pragma unroll
    for (int j = 0; j < 3; j++) {
        const float* te = task + ((size_t)b * NT_ + nb[j]) * H_;
        #pragma unroll
        for (int h = 0; h < H_; h++) out[h] += w[j] * te[h];
    }
    #pragma unroll
    for (int h = 0; h < H_; h++) agg_nbr[(size_t)i * H_ + h] = out[h];
}

// ---------------------------------------------------------------------------
// Kernel 4: uk = usv_new @ Wk + bk
// ---------------------------------------------------------------------------
__global__ __launch_bounds__(256) void hgnn_uk_kernel(
    const float* __restrict__ usv_new, const float* __restrict__ Wk,
    const float* __restrict__ bk, float* __restrict__ uk)
{
    int i = blockIdx.x * blockDim.x + threadIdx.x;
    if (i >= B_ * NU_) return;
    float xv[H_];
    #pragma unroll
    for (int h = 0; h < H_; h++) xv[h] = usv_new[(size_t)i * H_ + h];
    #pragma unroll
    for (int o = 0; o < H_; o++) {
        float s = bk[o];
        #pragma unroll
        for (int h = 0; h < H_; h++) s += xv[h] * Wk[h * H_ + o];
        uk[(size_t)i * H_ + o] = s;
    }
}

// ---------------------------------------------------------------------------
// Kernel 5: block 3 fused — task attends to usvs (transposed edge bias),
// WMMA aggregation over K=256, then ELU(task + agg_nbr + agg_usv) + LN.
// ---------------------------------------------------------------------------
__global__ __launch_bounds__(64) void hgnn_block3_kernel(
    const float* __restrict__ task,
    const int* __restrict__ adj, const float* __restrict__ ef,
    const float* __restrict__ We, const float* __restrict__ be,
    const float* __restrict__ Wau, const float* __restrict__ bau,
    const float* __restrict__ gnt_, const float* __restrict__ bnt_,
    const float* __restrict__ t_q, const float* __restrict__ uk,
    const float* __restrict__ usv_new, const float* __restrict__ agg_nbr,
    float* __restrict__ task_new)
{
    __shared__ float w_s[16][NU_];     // 16 KB
    __shared__ float uk_s[NU_][H_];    // 8 KB (row layout for score phase)
    __shared__ float un_t[H_][NU_];    // 8 KB (usv_new transposed, B matrix source)
    __shared__ float q_s[16][H_];
    __shared__ float agg_s[16][16];

    const int tid  = threadIdx.x;
    const int lane = tid & 31;
    const int wv   = tid >> 5;
    const int b    = blockIdx.x / (NT_ / 16);
    const int t0   = (blockIdx.x % (NT_ / 16)) * 16;

    float WeR[H_], beR[H_], WauR[H_];
    #pragma unroll
    for (int h = 0; h < H_; h++) { WeR[h] = We[h]; beR[h] = be[h]; WauR[h] = Wau[h]; }
    const float bau0 = bau[0];

    for (int i = tid; i < NU_ * H_; i += 64) {
        ((float*)uk_s)[i] = uk[(size_t)b * NU_ * H_ + i];
        float v = usv_new[(size_t)b * NU_ * H_ + i];
        int u = i >> 3, h = i & (H_ - 1);
        un_t[h][u] = v;
    }
    for (int i = tid; i < 16 * H_; i += 64)
        ((float*)q_s)[i] = t_q[((size_t)b * NT_ + t0) * H_ + i];
    __syncthreads();

    // ---- scores (transposed adjacency/edge access) ----
    for (int idx = tid; idx < 16 * NU_; idx += 64) {
        int r = idx >> 8, u = idx & (NU_ - 1);
        size_t g = ((size_t)(b * NU_ + u)) * NT_ + (t0 + r);
        float  e = ef[g];
        int    a = adj[g];
        float s = bau0;
        #pragma unroll
        for (int h = 0; h < H_; h++)
            s += tanhf(q_s[r][h] + uk_s[u][h] + e * WeR[h] + beR[h]) * WauR[h];
        w_s[r][u] = (a == 0) ? NEGV : s;
    }
    __syncthreads();

    // ---- softmax per row over NU ----
    for (int r = wv; r < 16; r += 2) {
        float m = -INFINITY;
        for (int u = lane; u < NU_; u += 32) m = fmaxf(m, w_s[r][u]);
        #pragma unroll
        for (int o = 16; o; o >>= 1) m = fmaxf(m, __shfl_xor(m, o, 32));
        float l = 0.f;
        for (int u = lane; u < NU_; u += 32) { float v = expf(w_s[r][u] - m); w_s[r][u] = v; l += v; }
        #pragma unroll
        for (int o = 16; o; o >>= 1) l += __shfl_xor(l, o, 32);
        float inv = 1.0f / l;
        for (int u = lane; u < NU_; u += 32) w_s[r][u] *= inv;
    }
    __syncthreads();

    // ---- aggregation via WMMA f32 16x16x4, K-loop over 256 ----
    if (wv == 0) {
        const int  half   = lane >> 4;
        const int  l16    = lane & 15;
        const bool nvalid = (l16 < H_);
        const int  bn     = l16 & (H_ - 1);
        v8f c = {0.f, 0.f, 0.f, 0.f, 0.f, 0.f, 0.f, 0.f};
        for (int k0 = 0; k0 < NU_; k0 += 4) {
            int ka = k0 + half * 2;
            float ax = w_s[l16][ka];
            float ay = w_s[l16][ka + 1];
            float tx = un_t[bn][ka];
            float ty = un_t[bn][ka + 1];
            v2f a, bb;
            a.x  = ax;
            a.y  = ay;
            bb.x = nvalid ? tx : 0.f;
            bb.y = nvalid ? ty : 0.f;
            c = __builtin_amdgcn_wmma_f32_16x16x4_f32(false, a, false, bb, (short)0, c, false, false);
        }
        #pragma unroll
        for (int j = 0; j < 8; j++) agg_s[half * 8 + j][l16] = c[j];
    }
    __syncthreads();

    // ---- ELU(task + agg_nbr + agg_usv) + LayerNorm -> task_new ----
    if (tid < 16) {
        int r = tid;
        size_t base = ((size_t)b * NT_ + t0 + r) * H_;
        float x[H_]; float mu = 0.f;
        #pragma unroll
        for (int h = 0; h < H_; h++) {
            float v = task[base + h] + agg_nbr[base + h] + agg_s[r][h];
            v = (v > 0.f) ? v : (expf(v) - 1.f);
            x[h] = v; mu += v;
        }
        mu *= (1.0f / H_);
        float var = 0.f;
        #pragma unroll
        for (int h = 0; h < H_; h++) { float d = x[h] - mu; var += d * d; }
        var *= (1.0f / H_);
        float rs = rsqrtf(var + EPS_);
        #pragma unroll
        for (int h = 0; h < H_; h++)
            task_new[base + h] = (x[h] - mu) * rs * gnt_[h] + bnt_[h];
    }
}

// ---------------------------------------------------------------------------
extern "C" void kernel_launch(void* const* d_in, const int* in_sizes, int n_in,
                              void* d_out, int out_size, void* d_ws, size_t ws_size,
                              hipStream_t stream)
{
    (void)in_sizes; (void)n_in; (void)out_size; (void)ws_size;

    const float* usv  = (const float*)d_in[0];
    const float* task = (const float*)d_in[1];
    const int*   adj  = (const int*)  d_in[2];
    const float* ef   = (const float*)d_in[3];
    const float* crd  = (const float*)d_in[4];
    const float* Wq   = (const float*)d_in[5];
    const float* bq   = (const float*)d_in[6];
    const float* Wk   = (const float*)d_in[7];
    const float* bk   = (const float*)d_in[8];
    // d_in[9] Wv, d_in[10] bv: computed-but-unused in the reference; skipped.
    const float* We   = (const float*)d_in[11];
    const float* be   = (const float*)d_in[12];
    const float* Wau  = (const float*)d_in[13];
    const float* bau  = (const float*)d_in[14];
    const float* Wat  = (const float*)d_in[15];
    const float* bat  = (const float*)d_in[16];
    const float* gnu_ = (const float*)d_in[17];
    const float* bnu_ = (const float*)d_in[18];
    const float* gnt_ = (const float*)d_in[19];
    const float* bnt_ = (const float*)d_in[20];

    float* usv_new  = (float*)d_out;                           // [B,NU,H]
    float* task_new = (float*)d_out + (size_t)B_ * NU_ * H_;   // [B,NT,H]

    // Workspace layout (floats): ~2.1 MB total
    float* ws      = (float*)d_ws;
    float* usv_q   = ws;                                   // B*NU*H
    float* task_k  = usv_q  + (size_t)B_ * NU_ * H_;       // B*NT*H
    float* t_q     = task_k + (size_t)B_ * NT_ * H_;       // B*NT*H
    float* uk      = t_q    + (size_t)B_ * NT_ * H_;       // B*NU*H
    float* agg_nbr = uk     + (size_t)B_ * NU_ * H_;       // B*NT*H

    hgnn_proj_kernel<<<(B_ * (NU_ + NT_) + 255) / 256, 256, 0, stream>>>(
        usv, task, Wq, bq, Wk, bk, usv_q, task_k, t_q);

    hgnn_block1_kernel<<<B_ * (NU_ / 16), 64, 0, stream>>>(
        usv, task, adj, ef, We, be, Wau, bau, gnu_, bnu_, usv_q, task_k, usv_new);

    hgnn_knn_kernel<<<(B_ * NT_ + 255) / 256, 256, 0, stream>>>(
        task, crd, t_q, task_k, Wat, bat, agg_nbr);

    hgnn_uk_kernel<<<(B_ * NU_ + 255) / 256, 256, 0, stream>>>(
        usv_new, Wk, bk, uk);

    hgnn_block3_kernel<<<B_ * (NT_ / 16), 64, 0, stream>>>(
        task, adj, ef, We, be, Wau, bau, gnt_, bnt_, t_q, uk, usv_new, agg_nbr, task_new);
}